// ReverseGeometryLoss_50912542327490
// MI455X (gfx1250) — compile-verified
//
#include <hip/hip_runtime.h>
#include <math.h>

#define B_ 32
#define N_ 2048
#define K_ 64
#define D_ 256

typedef __attribute__((ext_vector_type(2))) float v2f;
typedef __attribute__((ext_vector_type(8))) float v8f;

// ---------------------------------------------------------------------------
// Workspace layout (floats):
//   ws[0..31]   : per-batch ranking partial  (blockSum / max(n_non,1) * bvalid)
//   ws[32..63]  : per-batch count partial    (n_som * bvalid)
//   ws[64..95]  : per-batch state-KL partial
//   ws[96]      : diversity loss
//   ws[128 ..]  : normalized embeddings [64][256]
// ---------------------------------------------------------------------------

// ---- 1) Row-normalize state embeddings into ws scratch ---------------------
__global__ __launch_bounds__(256) void rgl_norm_kernel(
    const float* __restrict__ E, float* __restrict__ normed) {
  __shared__ float red[256];
  __shared__ float sh_inv;
  const int r = blockIdx.x;
  const int t = threadIdx.x;
  float x = E[r * D_ + t];
  red[t] = x * x;
  __syncthreads();
  for (int o = 128; o > 0; o >>= 1) {
    if (t < o) red[t] += red[t + o];
    __syncthreads();
  }
  if (t == 0) sh_inv = 1.0f / fmaxf(sqrtf(red[0]), 1e-12f);
  __syncthreads();
  normed[r * D_ + t] = x * sh_inv;
}

// ---- 2) Diversity: 64x64 Gram matrix via V_WMMA_F32_16X16X4_F32 ------------
// 16 waves; wave w owns tile (tr=w/4, tc=w%4) of sim = normed @ normed^T.
__global__ __launch_bounds__(512) void rgl_div_kernel(
    const float* __restrict__ normed, float* __restrict__ ws) {
  __shared__ float part[16];
  const int tid  = threadIdx.x;
  const int w    = tid >> 5;      // wave id 0..15
  const int lane = tid & 31;
  const int tr   = w >> 2;
  const int tc   = w & 3;
  // A-fragment (16x4 f32): lanes 0-15 hold K=k,k+1 ; lanes 16-31 hold K=k+2,k+3.
  // B-fragment (4x16 f32) mirrors it; both operands come from the same
  // row-major matrix, so the addressing pattern is identical modulo tile base.
  const int rowA = tr * 16 + (lane & 15);
  const int rowB = tc * 16 + (lane & 15);
  const int koff = (lane >> 4) << 1;   // 0 or 2
  const float* pa = normed + rowA * D_ + koff;
  const float* pb = normed + rowB * D_ + koff;

  v8f c = {};
#pragma unroll 4
  for (int kk = 0; kk < D_; kk += 4) {
    v2f a = *(const v2f*)(pa + kk);
    v2f b = *(const v2f*)(pb + kk);
    c = __builtin_amdgcn_wmma_f32_16x16x4_f32(
        /*neg_a=*/false, a, /*neg_b=*/false, b,
        /*c_mod=*/(short)0, c, /*reuse_a=*/false, /*reuse_b=*/false);
  }

  // Tile sum; subtract diagonal on diagonal tiles.
  // C/D layout: VGPR r, lanes 0-15 -> (M=r, N=lane); lanes 16-31 -> (M=r+8, N=lane-16).
  float tsum = 0.0f;
#pragma unroll
  for (int r = 0; r < 8; ++r) tsum += c[r];
  if (tr == tc) {
#pragma unroll
    for (int r = 0; r < 8; ++r) {
      if (lane == r || lane == r + 24) tsum -= c[r];
    }
  }
  // wave32 reduce
  for (int o = 16; o > 0; o >>= 1) tsum += __shfl_down(tsum, o, 32);
  if (lane == 0) part[w] = tsum;
  __syncthreads();
  if (tid == 0) {
    float tot = 0.0f;
    for (int i = 0; i < 16; ++i) tot += part[i];
    ws[96] = tot / (float)(K_ * (K_ - 1));
  }
}

// ---- 3) Ranking loss: one block per batch, scores/masks in LDS -------------
__global__ __launch_bounds__(256) void rgl_rank_kernel(
    const float* __restrict__ scores, const int* __restrict__ som_mask,
    const int* __restrict__ valid_mask, float* __restrict__ ws) {
  __shared__ float s_s[N_];
  __shared__ float s_non[N_];
  __shared__ float s_som[N_];
  __shared__ float red[256];
  __shared__ float sh_nNon, sh_nSom;
  const int b   = blockIdx.x;
  const int tid = threadIdx.x;

  float lnon = 0.0f, lsom = 0.0f;
  for (int n = tid; n < N_; n += 256) {
    float s = scores[b * N_ + n];
    int   v = valid_mask[b * N_ + n];
    int  sm = som_mask[b * N_ + n];
    float nf = (v != 0 && sm == 0) ? 1.0f : 0.0f;
    float sf = (v != 0 && sm != 0) ? 1.0f : 0.0f;
    s_s[n] = s; s_non[n] = nf; s_som[n] = sf;
    lnon += nf; lsom += sf;
  }
  red[tid] = lnon; __syncthreads();
  for (int o = 128; o > 0; o >>= 1) { if (tid < o) red[tid] += red[tid + o]; __syncthreads(); }
  if (tid == 0) sh_nNon = red[0];
  __syncthreads();
  red[tid] = lsom; __syncthreads();
  for (int o = 128; o > 0; o >>= 1) { if (tid < o) red[tid] += red[tid + o]; __syncthreads(); }
  if (tid == 0) sh_nSom = red[0];
  __syncthreads();

  // per_som denominator is batch-constant -> divide once after block reduce.
  float acc = 0.0f;
  for (int i = tid; i < N_; i += 256) {
    if (s_som[i] == 0.0f) continue;
    const float c1 = 1.0f - s_s[i];
    float a = 0.0f;
#pragma unroll 4
    for (int j = 0; j < N_; ++j) {
      a = fmaf(s_non[j], fmaxf(c1 + s_s[j], 0.0f), a);
    }
    acc += a;
  }
  red[tid] = acc; __syncthreads();
  for (int o = 128; o > 0; o >>= 1) { if (tid < o) red[tid] += red[tid + o]; __syncthreads(); }
  if (tid == 0) {
    const float nNon = sh_nNon, nSom = sh_nSom;
    const float bvalid = (nSom > 0.0f && nNon > 0.0f) ? 1.0f : 0.0f;
    ws[b]      = (red[0] / fmaxf(nNon, 1.0f)) * bvalid;
    ws[32 + b] = nSom * bvalid;
  }
}

// ---- 4) State KL: one 64-thread block per batch row ------------------------
__global__ __launch_bounds__(64) void rgl_state_kernel(
    const float* __restrict__ probs, const float* __restrict__ logits,
    float* __restrict__ ws) {
  __shared__ float red[64];
  const int b = blockIdx.x;
  const int t = threadIdx.x;
  const float l = logits[b * K_ + t];
  red[t] = l; __syncthreads();
  for (int o = 32; o > 0; o >>= 1) { if (t < o) red[t] = fmaxf(red[t], red[t + o]); __syncthreads(); }
  const float m = red[0];
  __syncthreads();
  const float e = expf(l - m);
  red[t] = e; __syncthreads();
  for (int o = 32; o > 0; o >>= 1) { if (t < o) red[t] += red[t + o]; __syncthreads(); }
  const float Z = red[0];
  __syncthreads();
  const float tp   = e / Z;
  const float logt = (l - m) - logf(Z);
  const float term = tp * (logt - logf(probs[b * K_ + t] + 1e-8f));
  red[t] = term; __syncthreads();
  for (int o = 32; o > 0; o >>= 1) { if (t < o) red[t] += red[t + o]; __syncthreads(); }
  if (t == 0) ws[64 + b] = red[0];
}

// ---- 5) Deterministic fixed-order combine ----------------------------------
__global__ void rgl_final_kernel(const float* __restrict__ ws,
                                 float* __restrict__ out) {
  if (threadIdx.x == 0 && blockIdx.x == 0) {
    float rs = 0.0f, cnt = 0.0f, ss = 0.0f;
    for (int i = 0; i < 32; ++i) { rs += ws[i]; cnt += ws[32 + i]; ss += ws[64 + i]; }
    const float rl = (cnt > 0.0f) ? rs / fmaxf(cnt, 1.0f) : 0.0f;
    out[0] = 1.0f * rl + 0.5f * (ss / (float)B_) + 0.1f * ws[96];
  }
}

extern "C" void kernel_launch(void* const* d_in, const int* in_sizes, int n_in,
                              void* d_out, int out_size, void* d_ws, size_t ws_size,
                              hipStream_t stream) {
  const float* scores = (const float*)d_in[0];
  const float* probs  = (const float*)d_in[1];
  const float* logits = (const float*)d_in[2];
  const float* emb    = (const float*)d_in[3];
  const int*   som    = (const int*)d_in[4];
  const int*   valid  = (const int*)d_in[5];
  float* ws     = (float*)d_ws;
  float* normed = ws + 128;

  rgl_norm_kernel<<<K_, D_, 0, stream>>>(emb, normed);
  rgl_div_kernel<<<1, 512, 0, stream>>>(normed, ws);
  rgl_rank_kernel<<<B_, 256, 0, stream>>>(scores, som, valid, ws);
  rgl_state_kernel<<<B_, K_, 0, stream>>>(probs, logits, ws);
  rgl_final_kernel<<<1, 32, 0, stream>>>(ws, (float*)d_out);
}